// MyModel_87522843560476
// MI455X (gfx1250) — compile-verified
//
#include <hip/hip_runtime.h>

// Batched scatter-add densification (SparseTensor -> dense), MI455X / gfx1250.
//
// out[b,h,w] = sum over valid p of feats[b,p] where indices[b,p] == (h,w)
// B=256 batches, P=4096 points, plane 512x512 f32.
//
// Memory-bound: ~268 MB output fill dominates (~12 us @ 23.3 TB/s HBM).
// Design:
//  - one 1024-thread block (32 wave32 waves) per batch plane (disjoint 1 MB)
//  - preload this thread's 4 points (indices+feature) with NT loads, issued
//    FIRST so their latency hides under the store clause + fence
//  - zero the plane with a clause of 64 coalesced global_store_b128
//  - __threadfence + __syncthreads: stores reach L2 (WGP$ is read-only on
//    CDNA5, atomics execute at L2) before any lane of this block scatters
//  - scatter-add with native non-returning global_atomic_add_f32
//    (relaxed, agent scope); zeroed lines are RT-resident in the 192 MB L2,
//    so the ~1M random atomics mostly hit L2

#define BATCH 256
#define NPTS  4096
#define HDIM  512
#define WDIM  512
#define PLANE (HDIM * WDIM)      // 262144 floats = 1 MB
#define PLANE4 (PLANE / 4)       // 65536 float4
#define TPB   1024               // 32 waves (wave32)
#define PPT   (NPTS / TPB)       // 4 points per thread

__global__ __launch_bounds__(TPB) void scatter_densify_kernel(
    const int*   __restrict__ indices,    // [B, P, 2] int32 (h, w)
    const int*   __restrict__ num_valid,  // [B] int32
    const float* __restrict__ feats,      // [B, P] float32
    float*       __restrict__ out)        // [B, H, W] float32
{
    const int b = blockIdx.x;
    const int t = threadIdx.x;

    float* plane = out + (size_t)b * PLANE;

    // ---- issue point loads first (single-use -> non-temporal) ----
    const int nv   = num_valid[b];                  // scalar load (kmcnt)
    const int base = b * NPTS;
    const long long* __restrict__ idx64 =
        reinterpret_cast<const long long*>(indices);  // (h,w) packed: h = low 32

    long long hw[PPT];
    float     v[PPT];
#pragma unroll
    for (int i = 0; i < PPT; ++i) {
        const int p = t + i * TPB;                  // always < NPTS: safe load
        hw[i] = __builtin_nontemporal_load(idx64 + base + p);   // b64, th:NT
        v[i]  = __builtin_nontemporal_load(feats + base + p);   // b32, th:NT
    }

    // ---- zero this batch's plane with b128 stores ----
    {
        float4* plane4 = reinterpret_cast<float4*>(plane);
        const float4 z = make_float4(0.f, 0.f, 0.f, 0.f);
#pragma unroll
        for (int i = 0; i < PLANE4 / TPB; ++i) {    // 64 coalesced b128 stores
            plane4[i * TPB + t] = z;
        }
    }

    // Stores must be visible at L2 (agent scope) before any lane of this
    // block issues atomics to the same plane.
    __threadfence();
    __syncthreads();

    // ---- scatter-add valid points ----
#pragma unroll
    for (int i = 0; i < PPT; ++i) {
        const int p = t + i * TPB;
        if (p < nv) {                               // padded points dropped
            const int h = (int)(hw[i] & 0xffffffffll);
            const int w = (int)(hw[i] >> 32);
            // valid points are always in [0,512)^2 by construction
            __hip_atomic_fetch_add(&plane[h * WDIM + w], v[i],
                                   __ATOMIC_RELAXED, __HIP_MEMORY_SCOPE_AGENT);
        }
    }
}

extern "C" void kernel_launch(void* const* d_in, const int* in_sizes, int n_in,
                              void* d_out, int out_size, void* d_ws, size_t ws_size,
                              hipStream_t stream) {
    // setup_inputs() order: indices [B,P,2] int, num_valid [B] int, feats [B,P,1] f32
    const int*   indices   = (const int*)  d_in[0];
    const int*   num_valid = (const int*)  d_in[1];
    const float* feats     = (const float*)d_in[2];
    float*       out       = (float*)      d_out;   // [B,H,W] f32

    scatter_densify_kernel<<<dim3(BATCH), dim3(TPB), 0, stream>>>(
        indices, num_valid, feats, out);
}